// GIN_88648124991293
// MI455X (gfx1250) — compile-verified
//
#include <hip/hip_runtime.h>
#include <hip/hip_bf16.h>

typedef __attribute__((ext_vector_type(2))) float v2f;
typedef __attribute__((ext_vector_type(8))) float v8f;

#define N_NODES 100224
#define IN_DIM 116
#define HID 128
#define N_GRAPHS 864
#define BN_EPS 1e-5f

// ---------------- utility: zero a float region ----------------
__global__ void gin_zero(float* __restrict__ p, int n) {
    int i = blockIdx.x * blockDim.x + threadIdx.x;
    if (i < n) p[i] = 0.0f;
}

// ---------------- edge scatter-add: agg[dst] += x[src] ----------------
__global__ __launch_bounds__(128) void gin_scatter(const float* __restrict__ x,
                                                   const int* __restrict__ ei,
                                                   float* __restrict__ agg,
                                                   int nEdges) {
    int e = blockIdx.x;
    int f = threadIdx.x;
    if (f >= IN_DIM) return;
    int s = ei[e];
    int d = ei[nEdges + e];
    atomicAdd(&agg[(size_t)d * IN_DIM + f], x[(size_t)s * IN_DIM + f]);
}

// ---------------- h0 = x + agg (in place into agg) ----------------
__global__ void gin_addx(float* __restrict__ agg, const float* __restrict__ x, int n) {
    int i = blockIdx.x * blockDim.x + threadIdx.x;
    if (i < n) agg[i] += x[i];
}

// ---------------- per-graph node counts ----------------
__global__ void gin_counts(const int* __restrict__ batch, float* __restrict__ counts, int n) {
    int i = blockIdx.x * blockDim.x + threadIdx.x;
    if (i < n) atomicAdd(&counts[batch[i]], 1.0f);
}

// ---------------- GEMM1: h1 = relu(h0 @ W1 + b1), fp32 WMMA ----------------
// Each wave computes a 16x128 strip: one A fragment per K-step, reused by 8
// independent wmma ops (one per 16-col N-tile). M=100224 -> 6264 waves.
__global__ __launch_bounds__(128) void gin_gemm1(const float* __restrict__ h0,
                                                 const float* __restrict__ W1,
                                                 const float* __restrict__ b1,
                                                 float* __restrict__ h1) {
    const int mt   = blockIdx.x * 4 + threadIdx.y;   // M-tile (strip) index, uniform per wave
    const int lane = threadIdx.x;                    // 0..31 (wave32)
    const int hf   = lane >> 4;                      // half-wave: selects K pair
    const int l    = lane & 15;
    const int arow = mt * 16 + l;                    // A row streamed by this lane

    const float* __restrict__ aptr = h0 + (size_t)arow * IN_DIM;

    v8f acc[8] = {};
    for (int k0 = 0; k0 < IN_DIM; k0 += 4) {
        const int k = k0 + 2 * hf;                   // A layout: lanes 0-15 K=0,1; 16-31 K=2,3
        const v2f a = *(const v2f*)(aptr + k);       // 8B aligned; loaded ONCE per K-step
        const float* __restrict__ w0 = W1 + (size_t)k * HID + l;
        const float* __restrict__ w1 = w0 + HID;
#pragma unroll
        for (int nt = 0; nt < 8; ++nt) {
            v2f b;
            b.x = w0[nt * 16];
            b.y = w1[nt * 16];
            acc[nt] = __builtin_amdgcn_wmma_f32_16x16x4_f32(false, a, false, b,
                                                            (short)0, acc[nt], false, false);
        }
    }

#pragma unroll
    for (int nt = 0; nt < 8; ++nt) {
        const int col  = nt * 16 + l;
        const float bias = b1[col];
#pragma unroll
        for (int v = 0; v < 8; ++v) {
            const int orow = mt * 16 + v + 8 * hf;   // D layout: VGPR v -> M=v (+8 upper half)
            float r = acc[nt][v] + bias;
            r = r > 0.0f ? r : 0.0f;
            h1[(size_t)orow * HID + col] = r;
        }
    }
}

// ---------------- GEMM2 + ReLU + BN + ReLU + graph-sum scatter ----------------
// Same 16x128-strip structure, K=128 (32 steps). Epilogue fuses bias, ReLU,
// BatchNorm (eval), ReLU, and atomically accumulates into per-graph sums.
__global__ __launch_bounds__(128) void gin_gemm2(const float* __restrict__ h1,
                                                 const float* __restrict__ W2,
                                                 const float* __restrict__ b2,
                                                 const float* __restrict__ bn_gamma,
                                                 const float* __restrict__ bn_beta,
                                                 const float* __restrict__ bn_mean,
                                                 const float* __restrict__ bn_var,
                                                 const int* __restrict__ batch,
                                                 float* __restrict__ sums) {
    const int mt   = blockIdx.x * 4 + threadIdx.y;
    const int lane = threadIdx.x;
    const int hf   = lane >> 4;
    const int l    = lane & 15;
    const int arow = mt * 16 + l;

    const float* __restrict__ aptr = h1 + (size_t)arow * HID;

    v8f acc[8] = {};
    for (int k0 = 0; k0 < HID; k0 += 4) {
        const int k = k0 + 2 * hf;
        const v2f a = *(const v2f*)(aptr + k);
        const float* __restrict__ w0 = W2 + (size_t)k * HID + l;
        const float* __restrict__ w1 = w0 + HID;
#pragma unroll
        for (int nt = 0; nt < 8; ++nt) {
            v2f b;
            b.x = w0[nt * 16];
            b.y = w1[nt * 16];
            acc[nt] = __builtin_amdgcn_wmma_f32_16x16x4_f32(false, a, false, b,
                                                            (short)0, acc[nt], false, false);
        }
    }

    // per-row graph ids (row depends on v and hf only)
    int gid[8];
#pragma unroll
    for (int v = 0; v < 8; ++v) gid[v] = batch[mt * 16 + v + 8 * hf];

#pragma unroll
    for (int nt = 0; nt < 8; ++nt) {
        const int col   = nt * 16 + l;
        const float bias  = b2[col];
        const float scale = rsqrtf(bn_var[col] + BN_EPS) * bn_gamma[col];
        const float shift = bn_beta[col] - bn_mean[col] * scale;
#pragma unroll
        for (int v = 0; v < 8; ++v) {
            float r = acc[nt][v] + bias;
            r = r > 0.0f ? r : 0.0f;     // inner ReLU
            r = r * scale + shift;       // BatchNorm (eval mode)
            r = r > 0.0f ? r : 0.0f;     // outer ReLU
            atomicAdd(&sums[(size_t)gid[v] * HID + col], r);
        }
    }
}

// ---------------- pooled = sums/counts; x_t = pooled @ lin_W + lin_b ----------------
__global__ __launch_bounds__(128) void gin_pool(const float* __restrict__ sums,
                                                const float* __restrict__ counts,
                                                const float* __restrict__ lin_W,
                                                const float* __restrict__ lin_b,
                                                float* __restrict__ out) {
    const int gr = blockIdx.x;     // graph
    const int c  = threadIdx.x;    // channel 0..127
    float cnt = counts[gr];
    cnt = cnt > 1.0f ? cnt : 1.0f;
    const float p = sums[(size_t)gr * HID + c] / cnt;
    out[N_GRAPHS * 2 + (size_t)gr * HID + c] = p;   // pooled after x_t block

    __shared__ float r0[HID];
    __shared__ float r1[HID];
    r0[c] = p * lin_W[c * 2 + 0];
    r1[c] = p * lin_W[c * 2 + 1];
    __syncthreads();
#pragma unroll
    for (int s = HID / 2; s > 0; s >>= 1) {
        if (c < s) { r0[c] += r0[c + s]; r1[c] += r1[c + s]; }
        __syncthreads();
    }
    if (c == 0) {
        out[gr * 2 + 0] = r0[0] + lin_b[0];
        out[gr * 2 + 1] = r1[0] + lin_b[1];
    }
}

extern "C" void kernel_launch(void* const* d_in, const int* in_sizes, int n_in,
                              void* d_out, int out_size, void* d_ws, size_t ws_size,
                              hipStream_t stream) {
    const float* x        = (const float*)d_in[0];
    const int*   ei       = (const int*)d_in[1];
    const int*   batch    = (const int*)d_in[2];
    const float* W1       = (const float*)d_in[3];
    const float* b1       = (const float*)d_in[4];
    const float* W2       = (const float*)d_in[5];
    const float* b2       = (const float*)d_in[6];
    const float* bn_gamma = (const float*)d_in[7];
    const float* bn_beta  = (const float*)d_in[8];
    const float* bn_mean  = (const float*)d_in[9];
    const float* bn_var   = (const float*)d_in[10];
    const float* lin_W    = (const float*)d_in[11];
    const float* lin_b    = (const float*)d_in[12];
    float* out = (float*)d_out;

    const int nEdges = in_sizes[1] / 2;

    // workspace layout (floats)
    const size_t AGG_N  = (size_t)N_NODES * IN_DIM;   // 11,625,984
    const size_t H1_N   = (size_t)N_NODES * HID;      // 12,828,672
    const size_t SUMS_N = (size_t)N_GRAPHS * HID;     // 110,592
    float* agg    = (float*)d_ws;
    float* h1     = agg + AGG_N;
    float* sums   = h1 + H1_N;
    float* counts = sums + SUMS_N;                    // 864

    // 1) zero accumulators (agg, then sums+counts contiguous)
    gin_zero<<<(int)((AGG_N + 255) / 256), 256, 0, stream>>>(agg, (int)AGG_N);
    gin_zero<<<(int)((SUMS_N + N_GRAPHS + 255) / 256), 256, 0, stream>>>(sums, (int)(SUMS_N + N_GRAPHS));

    // 2) edge scatter-add (L2-resident agg)
    gin_scatter<<<nEdges, 128, 0, stream>>>(x, ei, agg, nEdges);

    // 3) h0 = x + agg (in place)
    gin_addx<<<(int)((AGG_N + 255) / 256), 256, 0, stream>>>(agg, x, (int)AGG_N);

    // 4) per-graph counts
    gin_counts<<<(N_NODES + 255) / 256, 256, 0, stream>>>(batch, counts, N_NODES);

    // 5) GEMM1 (fp32 WMMA, 16x128 strips): 6264 waves, 4 waves/block
    dim3 gblk(32, 4);
    const int strips = N_NODES / 16;                  // 6264
    gin_gemm1<<<strips / 4, gblk, 0, stream>>>(agg, W1, b1, h1);

    // 6) GEMM2 + BN + ReLU + graph-sum scatter
    gin_gemm2<<<strips / 4, gblk, 0, stream>>>(h1, W2, b2, bn_gamma, bn_beta,
                                               bn_mean, bn_var, batch, sums);

    // 7) mean pool + linear head
    gin_pool<<<N_GRAPHS, HID, 0, stream>>>(sums, counts, lin_W, lin_b, out);
}